// GConv_30829275251152
// MI455X (gfx1250) — compile-verified
//
#include <hip/hip_runtime.h>

// GConv as batched GEMM on CDNA5 WMMA (fp32, V_WMMA_F32_16X16X4_F32).
// out[b,o,g,s] = sum_{c,n} Psi[o,c,GnnPerms[g,n]] * In[b,c,NNsites[n,s]] + bias[o]
// M = (o,g) = 384, K = (c,n) = 104, N = s = 512, batched over b = 512.
//
// Round-2 changes:
//  * W_all (384x104) precomputed once into d_ws -> A fragments become single
//    b64 loads (row-major (kk,hi,t) order == plain k order), no /13 math.
//  * Gather staging uses async global->LDS (global_load_async_to_lds_b32,
//    GVS mode), fenced with s_wait_asynccnt before the barrier.

typedef __attribute__((ext_vector_type(2))) float v2f;   // A/B fragment
typedef __attribute__((ext_vector_type(8))) float v8f;   // C/D accumulator

#define NB_B    512
#define C_IN    8
#define C_OUT   8
#define N_NGB   13
#define NG      48
#define NSITES  512
#define KDIM    (C_IN * N_NGB)     // 104
#define KSTEPS  (KDIM / 4)         // 26
#define MROWS   (C_OUT * NG)       // 384
#define SSTRIP  128
#define NTILES  (SSTRIP / 16)      // 8
#define GSTRIDE 136                // LDS row stride (floats): half-wave rows hit
                                   // disjoint bank ranges

// ---- Prep: W_all[r][k] = Psi[o, c, GnnPerms[g, n]],  r=o*48+g, k=c*13+n ----
__global__ __launch_bounds__(256)
void build_wpack(const float* __restrict__ Psi,
                 const int*   __restrict__ GnnPerms,
                 float*       __restrict__ Wpack)
{
    int idx = blockIdx.x * 256 + threadIdx.x;
    if (idx >= MROWS * KDIM) return;
    int r  = idx / KDIM;
    int k  = idx - r * KDIM;
    int o  = r / NG;
    int g  = r - o * NG;
    int c  = k / N_NGB;
    int nb = k - c * N_NGB;
    Wpack[idx] = Psi[(o * C_IN + c) * N_NGB + GnnPerms[g * N_NGB + nb]];
}

__global__ __launch_bounds__(256, 1)
void gconv_wmma_f32(const float* __restrict__ In,        // [B, C_IN, NSITES]
                    const float* __restrict__ Wpack,     // [MROWS, KDIM]
                    const float* __restrict__ bias,      // [C_OUT]
                    const int*   __restrict__ NNsites,   // [N_NGB, NSITES]
                    float*       __restrict__ Out)       // [B, C_OUT, NG, NSITES]
{
    __shared__ float G_s[KDIM * GSTRIDE];                // 56.6 KB

    const int tid = threadIdx.x;
    const int b   = blockIdx.y;
    const int s0  = blockIdx.x * SSTRIP;

    const float* __restrict__ Inb = In + (size_t)b * (C_IN * NSITES);
    const unsigned lds_base = (unsigned)(size_t)(&G_s[0]);   // LDS byte offset

    // ---- Async gather into LDS: G_s[k][scol] = In[b, c, NNsites[nb, s0+scol]].
    //      Per-lane: 32-bit global offset from SGPR base Inb (GVS mode),
    //      per-lane LDS destination address. Tracked by ASYNCcnt.
    #pragma unroll 4
    for (int it = 0; it < (KDIM * SSTRIP) / 256; ++it) { // 52 iters
        int e    = tid + it * 256;
        int k    = e >> 7;                               // / 128
        int scol = e & 127;
        int c    = (int)((unsigned)k / N_NGB);
        int nb   = k - c * N_NGB;
        int site = NNsites[nb * NSITES + s0 + scol];
        unsigned goff = (unsigned)((c * NSITES + site) * 4);        // < 16 KB
        unsigned loff = lds_base + (unsigned)((k * GSTRIDE + scol) * 4);
        asm volatile("global_load_async_to_lds_b32 %0, %1, %2"
                     :
                     : "v"(loff), "v"(goff), "s"(Inb)
                     : "memory");
    }

    // ---- Per-wave role: wave w == output channel o (8 waves = C_OUT).
    const int wave = tid >> 5;        // o
    const int lane = tid & 31;
    const int lrow = lane & 15;       // A: M row / B,D: N col
    const int hi   = lane >> 4;       // K-pair select (A/B) / +8 row (D)

    // ---- Resident A fragments: one b64 load each from row-major W_all.
    //      a[i][kk] = W_all[m][4kk+2hi .. 4kk+2hi+1], m = wave*48 + i*16 + lrow.
    v2f a[3][KSTEPS];
    #pragma unroll
    for (int i = 0; i < 3; ++i) {
        const int m = wave * NG + i * 16 + lrow;
        const v2f* __restrict__ wrow = (const v2f*)(Wpack + (size_t)m * KDIM);
        #pragma unroll
        for (int kk = 0; kk < KSTEPS; ++kk)
            a[i][kk] = wrow[kk * 2 + hi];
    }

    const float bv = bias[wave];

    // Async-LDS writes must land before any wave reads G_s.
    asm volatile("s_wait_asynccnt 0" ::: "memory");
    __syncthreads();

    float* __restrict__ outb = Out + (size_t)b * (MROWS * NSITES);

    // ---- 8 N-tiles; per tile: 26 k-steps x 3 WMMAs (independent accumulators).
    for (int nt = 0; nt < NTILES; ++nt) {
        v8f acc[3] = {};
        const int colbase = nt * 16 + lrow;
        #pragma unroll
        for (int kk = 0; kk < KSTEPS; ++kk) {
            const int k0 = kk * 4 + hi * 2;
            v2f bf;                                      // B (4x16): mirror of A
            bf.x = G_s[k0 * GSTRIDE + colbase];
            bf.y = G_s[(k0 + 1) * GSTRIDE + colbase];
            #pragma unroll
            for (int i = 0; i < 3; ++i)
                acc[i] = __builtin_amdgcn_wmma_f32_16x16x4_f32(
                    false, a[i][kk], false, bf, (short)0, acc[i], false, false);
        }
        // D layout: VGPR v -> row (v + 8*hi), col = lane&15; r = o*48+g matches
        // the contiguous [o,g] output layout.
        const int s = s0 + nt * 16 + lrow;
        #pragma unroll
        for (int i = 0; i < 3; ++i) {
            #pragma unroll
            for (int v = 0; v < 8; ++v) {
                int r = wave * NG + i * 16 + hi * 8 + v;
                outb[r * NSITES + s] = acc[i][v] + bv;
            }
        }
    }
}

extern "C" void kernel_launch(void* const* d_in, const int* in_sizes, int n_in,
                              void* d_out, int out_size, void* d_ws, size_t ws_size,
                              hipStream_t stream) {
    (void)in_sizes; (void)n_in; (void)out_size; (void)ws_size;
    const float* In       = (const float*)d_in[0];
    const float* Psi      = (const float*)d_in[1];
    const float* bias     = (const float*)d_in[2];
    const int*   GnnPerms = (const int*)d_in[3];
    const int*   NNsites  = (const int*)d_in[4];
    float*       Out      = (float*)d_out;
    float*       Wpack    = (float*)d_ws;               // 384*104*4 = 159,744 B

    // 1) Build permuted weight matrix once (same for all b).
    build_wpack<<<(MROWS * KDIM + 255) / 256, 256, 0, stream>>>(Psi, GnnPerms, Wpack);

    // 2) Main batched WMMA GEMM.
    dim3 grid(NSITES / SSTRIP, NB_B);   // (4, 512)
    gconv_wmma_f32<<<grid, 256, 0, stream>>>(In, Wpack, bias, NNsites, Out);
}